// PermutationClosedStructure_37486474560133
// MI455X (gfx1250) — compile-verified
//
#include <hip/hip_runtime.h>

typedef __attribute__((ext_vector_type(2))) float v2f;
typedef __attribute__((ext_vector_type(4))) float v4f;
typedef __attribute__((ext_vector_type(8))) float v8f;

#define B_    128
#define L_    9
#define CIN_  256
#define COUT_ 256
#define K_    3
#define P_    1680
#define MROWS (B_*L_)      // 1152  (rows of Z: r = b*L + l)
#define NCOLS (K_*COUT_)   // 768   (cols of Z: j = a*COUT + o)
#define PPB   240          // permutations per block in scatter kernel (1680/7)

// ---------------------------------------------------------------------------
// Kernel 0: decode one-hot mask M (K,P,L) into packed base-3 codes.
// code[p] bits[2l+1:2l] = argmax_a M[a,p,l]  (exactly one a is 1.0)
// ---------------------------------------------------------------------------
__global__ void build_codes(const float* __restrict__ M, unsigned* __restrict__ codes) {
    int p = blockIdx.x * blockDim.x + threadIdx.x;
    if (p >= P_) return;
    unsigned code = 0;
    #pragma unroll
    for (int l = 0; l < L_; ++l) {
        float a1 = M[(size_t)1 * P_ * L_ + (size_t)p * L_ + l];
        float a2 = M[(size_t)2 * P_ * L_ + (size_t)p * L_ + l];
        unsigned a = (unsigned)(a1 + 2.0f * a2 + 0.5f);   // 0,1,2
        code |= a << (2 * l);
    }
    codes[p] = code;
}

// ---------------------------------------------------------------------------
// Kernel 1: Z = X * Wflat^T via V_WMMA_F32_16X16X4_F32 (full f32 fidelity).
//   X     : (1152, 256) row-major   (= x reshaped (B*L, C_IN))
//   Wflat : (768, 256)  row-major   (= W reshaped (K*C_OUT, C_IN))
//   Z     : (1152, 768) row-major, Z[b*9+l][a*256+o]
// One wave per 16x16 output tile; K loop of 64 WMMA steps.
// A layout (32b 16x4): lanes0-15 M=lane K={0,1}; lanes16-31 M=lane-16 K={2,3}
// ---------------------------------------------------------------------------
__global__ void __launch_bounds__(256)
gemm_z(const float* __restrict__ X, const float* __restrict__ W,
       float* __restrict__ Z) {
    const int lane  = threadIdx.x & 31;
    const int wid   = blockIdx.x * (blockDim.x >> 5) + (threadIdx.x >> 5);
    const int mtile = wid / (NCOLS / 16);     // 0..71
    const int ntile = wid % (NCOLS / 16);     // 0..47
    const int lrow  = lane & 15;
    const int khalf = (lane < 16) ? 0 : 2;

    const float* arow = X + (size_t)(mtile * 16 + lrow) * CIN_ + khalf;
    const float* brow = W + (size_t)(ntile * 16 + lrow) * CIN_ + khalf;

    v8f c = {};
    #pragma unroll 8
    for (int k = 0; k < CIN_; k += 4) {
        v2f a = *(const v2f*)(arow + k);   // 8B-aligned (k+khalf even)
        v2f b = *(const v2f*)(brow + k);
        c = __builtin_amdgcn_wmma_f32_16x16x4_f32(
                /*neg_a=*/false, a, /*neg_b=*/false, b,
                /*c_mod=*/(short)0, c, /*reuse_a=*/false, /*reuse_b=*/false);
    }

    // C/D layout: VGPR v -> M = v + (lane<16 ? 0 : 8), N = lane%16
    const int n     = ntile * 16 + lrow;
    const int mbase = mtile * 16 + ((lane < 16) ? 0 : 8);
    #pragma unroll
    for (int v = 0; v < 8; ++v)
        Z[(size_t)(mbase + v) * NCOLS + n] = c[v];
}

// ---------------------------------------------------------------------------
// Kernel 2: y[b,p,o] = sum_l Z[b, l, code[p][l], o]
// Grid (7, 128): block = (b, 240 permutations). Z[b] (27KB) staged in LDS.
// Each wave owns one p per iteration (code uniform per wave); 16B-vector LDS
// reads, coalesced non-temporal 16B stores (y is 220MB write-once >> L2).
// ---------------------------------------------------------------------------
__global__ void __launch_bounds__(256)
scatter_y(const float* __restrict__ Z, const unsigned* __restrict__ codes,
          float* __restrict__ y) {
    __shared__ __align__(16) float    zb[L_ * K_ * COUT_];  // 6912 f32 = 27KB
    __shared__ unsigned pc[PPB];

    const int tid   = threadIdx.x;
    const int b     = blockIdx.y;
    const int pbase = blockIdx.x * PPB;

    // stage Z[b] slab (rows b*9 .. b*9+8, contiguous 6912 floats)
    const v4f* zg = (const v4f*)(Z + (size_t)b * L_ * NCOLS);
    v4f* zs = (v4f*)zb;
    for (int i = tid; i < (L_ * K_ * COUT_ / 4); i += 256) zs[i] = zg[i];
    for (int i = tid; i < PPB; i += 256) pc[i] = codes[pbase + i];
    __syncthreads();

    const int po = tid >> 6;          // 0..3 : which of 4 concurrent p rows
    const int o4 = (tid & 63) << 2;   // output channel group (float4)

    for (int it = 0; it < PPB / 4; ++it) {
        const int pl = it * 4 + po;
        const unsigned code = pc[pl];
        v4f acc = {0.f, 0.f, 0.f, 0.f};
        #pragma unroll
        for (int l = 0; l < L_; ++l) {
            const int a = (code >> (2 * l)) & 3;
            const v4f zv = *(const v4f*)(zb + (l * 3 + a) * COUT_ + o4);
            acc += zv;
        }
        v4f* dst = (v4f*)(y + ((size_t)b * P_ + (pbase + pl)) * COUT_ + o4);
        __builtin_nontemporal_store(acc, dst);
    }
}

// ---------------------------------------------------------------------------
extern "C" void kernel_launch(void* const* d_in, const int* in_sizes, int n_in,
                              void* d_out, int out_size, void* d_ws, size_t ws_size,
                              hipStream_t stream) {
    const float* x = (const float*)d_in[0];   // (B, L, C_IN)
    const float* W = (const float*)d_in[1];   // (K, C_OUT, C_IN)
    const float* M = (const float*)d_in[2];   // (K, P, L)
    float* y = (float*)d_out;                 // (B, P, C_OUT)

    float*    Z     = (float*)d_ws;                               // 3.54 MB
    unsigned* codes = (unsigned*)((char*)d_ws +
                        (size_t)MROWS * NCOLS * sizeof(float));   // +6.7 KB

    build_codes<<<(P_ + 255) / 256, 256, 0, stream>>>(M, codes);

    // 72 * 48 = 3456 tiles, 8 waves/block -> 432 blocks
    gemm_z<<<(MROWS / 16) * (NCOLS / 16) / 8, 256, 0, stream>>>(x, W, Z);

    scatter_y<<<dim3(P_ / PPB, B_), 256, 0, stream>>>(Z, codes, y);
}